// GradientCalculation_CP_Delaunay_weight_21852793602447
// MI455X (gfx1250) — compile-verified
//
#include <hip/hip_runtime.h>
#include <hip/hip_bf16.h>

#define BATCH 2
#define NPTS 8192
#define KNN 6
#define G 5
#define TWO_PI_F 6.28318530717958647692f

typedef float v2f __attribute__((ext_vector_type(2)));
typedef float v8f __attribute__((ext_vector_type(8)));

// ---------------------------------------------------------------------------
// Kernel 1: KNN via V_WMMA_F32_16X16X4_F32.
// One wave (32 lanes) owns 16 query points. For each 16-candidate tile one
// WMMA produces the full 16x16 squared-distance tile via the embedding
//   A row m = (-2*xc, -2*yc, 1, |c|^2),  B col n = (xq, yq, |q|^2, 1)
// so D[m][n] = |c-q|^2 exactly (expanded form).
// D layout: lane l holds column N = l&15, rows M = r + 8*(l>>4), r=0..7.
// Top-k is branch-free: (dist,idx) packed into a u64 key (dist >= 0 so IEEE
// bits order monotonically as unsigned); each insert level lowers to
// v_min_u64 + v_max_u64. A wave-uniform early-out skips the insert chain for
// tiles where no lane's candidates can enter its top-6 (most tiles, since
// expected inserts per lane ~ 6*ln(4096) ~ 50 over the whole 4096-cand scan).
// Lane halves merge through LDS at the end.
// ---------------------------------------------------------------------------
__global__ __launch_bounds__(32)
void knn_wmma_kernel(const float* __restrict__ coord,
                     int* __restrict__ idx_ws,
                     float* __restrict__ idx_out)
{
    const int wave  = blockIdx.x;            // 0 .. B*N/16-1
    const int b     = wave / (NPTS / 16);
    const int qTile = wave % (NPTS / 16);
    const int lane  = threadIdx.x;
    const int col   = lane & 15;
    const int half  = lane >> 4;
    const float* cb = coord + (size_t)b * NPTS * 2;

    // B fragment (queries): VGPR0 = rows K0 (lanes 0-15) / K1 (lanes 16-31),
    //                       VGPR1 = rows K2 (lanes 0-15) / K3 (lanes 16-31).
    const int   q  = qTile * 16 + col;
    const float qx = cb[2 * q + 0];
    const float qy = cb[2 * q + 1];
    const float qn = qx * qx + qy * qy;
    v2f bfrag;
    bfrag.x = half ? qy   : qx;    // K0 = xq, K1 = yq
    bfrag.y = half ? 1.0f : qn;    // K2 = |q|^2, K3 = 1

    // sorted ascending top-6 as u64 keys: (f32 dist bits << 32) | index
    unsigned long long lk[KNN];
#pragma unroll
    for (int j = 0; j < KNN; ++j) lk[j] = 0xFFFFFFFFFFFFFFFFull;

    for (int t = 0; t < NPTS / 16; ++t) {
        // A fragment (candidates): lanes 0-15 hold (K0,K1) for row m=lane,
        // lanes 16-31 hold (K2,K3) for row m=lane-16.
        const int   c0 = t * 16 + col;
        const float cx = cb[2 * c0 + 0];
        const float cy = cb[2 * c0 + 1];
        const float cn = cx * cx + cy * cy;
        v2f afrag;
        afrag.x = half ? 1.0f : (-2.0f * cx);   // K0 = -2xc, K2 = 1
        afrag.y = half ? cn   : (-2.0f * cy);   // K1 = -2yc, K3 = |c|^2

        v8f acc = {0.f, 0.f, 0.f, 0.f, 0.f, 0.f, 0.f, 0.f};
        acc = __builtin_amdgcn_wmma_f32_16x16x4_f32(
                  false, afrag, false, bfrag, (short)0, acc, false, false);

        // Wave-uniform early-out: can any of this lane's 8 candidates enter
        // its top-6?  Compare in the unsigned-bits domain so the 0xFFFFFFFF
        // sentinel (NaN pattern) of a not-yet-full list always admits.
        float dmin = acc[0];
#pragma unroll
        for (int r = 1; r < 8; ++r) dmin = fminf(dmin, acc[r]);
        const unsigned int dminBits = __float_as_uint(fmaxf(dmin, 0.0f));
        const unsigned int worstHi  = (unsigned int)(lk[KNN - 1] >> 32);
        if (__any(dminBits <= worstHi)) {
            const int mbase = t * 16 + half * 8;
#pragma unroll
            for (int r = 0; r < 8; ++r) {
                const float dd = fmaxf(acc[r], 0.0f);   // dist >= 0 for key order
                unsigned long long key =
                    ((unsigned long long)__float_as_uint(dd) << 32) |
                    (unsigned int)(mbase + r);
                // branch-free sorted insert (ascending); ties -> lower index
#pragma unroll
                for (int j = 0; j < KNN; ++j) {
                    const unsigned long long mn = key < lk[j] ? key   : lk[j];
                    const unsigned long long mx = key < lk[j] ? lk[j] : key;
                    lk[j] = mn;
                    key   = mx;
                }
            }
        }
    }

    // Merge the two lanes that share a query column.
    __shared__ unsigned long long skey[32 * KNN];
#pragma unroll
    for (int j = 0; j < KNN; ++j) skey[lane * KNN + j] = lk[j];
    __syncthreads();

    if (lane < 16) {
        const unsigned long long* ka = &skey[lane * KNN];
        const unsigned long long* kb = &skey[(lane + 16) * KNN];
        int a = 0, c = 0;
        const size_t base = ((size_t)b * NPTS + (size_t)(qTile * 16 + lane)) * KNN;
#pragma unroll
        for (int k = 0; k < KNN; ++k) {
            const bool ta = ka[a] < kb[c];
            const unsigned long long sel = ta ? ka[a] : kb[c];
            if (ta) ++a; else ++c;
            const int sidx = (int)(unsigned int)(sel & 0xFFFFFFFFu);
            idx_ws[base + k]  = sidx;
            idx_out[base + k] = (float)sidx;
        }
    }
}

// ---------------------------------------------------------------------------
// Kernel 2: umbrella features, one thread per point.
// ---------------------------------------------------------------------------
__global__ __launch_bounds__(256)
void umbrella_kernel(const float* __restrict__ coord,
                     const float* __restrict__ value,
                     const int*   __restrict__ idx_ws,
                     float* __restrict__ grad_out,
                     float* __restrict__ umb_out,
                     float* __restrict__ sidx_out)
{
    const int gid = blockIdx.x * blockDim.x + threadIdx.x;
    if (gid >= BATCH * NPTS) return;
    const int b = gid / NPTS;
    const int n = gid - b * NPTS;
    const float* cb = coord + (size_t)b * NPTS * 2;
    const float* vb = value + (size_t)b * NPTS * 3;

    const float px = cb[2 * n + 0];
    const float py = cb[2 * n + 1];
    float pv[3] = { vb[3 * n + 0], vb[3 * n + 1], vb[3 * n + 2] };

    float ux[G], uy[G], uv[G][3];     // uncentered neighbors
    float rx[G], ry[G], rv[G][3];     // centered neighbors
    float ph[G];
#pragma unroll
    for (int j = 0; j < G; ++j) {
        const int id = idx_ws[(size_t)gid * KNN + j + 1];   // skip self
        ux[j] = cb[2 * id + 0];
        uy[j] = cb[2 * id + 1];
        uv[j][0] = vb[3 * id + 0];
        uv[j][1] = vb[3 * id + 1];
        uv[j][2] = vb[3 * id + 2];
        rx[j] = ux[j] - px;
        ry[j] = uy[j] - py;
        rv[j][0] = uv[j][0] - pv[0];
        rv[j][1] = uv[j][1] - pv[1];
        rv[j][2] = uv[j][2] - pv[2];
        ph[j] = atan2f(ry[j], rx[j]) / TWO_PI_F + 0.5f;
    }

    // stable insertion sort on phi (matches jnp.argsort)
    int   so[G] = {0, 1, 2, 3, 4};
    float ps[G];
#pragma unroll
    for (int j = 0; j < G; ++j) ps[j] = ph[j];
    for (int i = 1; i < G; ++i) {
        const float kp = ps[i];
        const int   ks = so[i];
        int j = i - 1;
        while (j >= 0 && ps[j] > kp) { ps[j + 1] = ps[j]; so[j + 1] = so[j]; --j; }
        ps[j + 1] = kp;
        so[j + 1] = ks;
    }

    // triangle areas and weights
    float area[G], asum = 0.f;
#pragma unroll
    for (int j = 0; j < G; ++j) {
        const int s0 = so[j], s1 = so[(j + 1) % G];
        const float a = 0.5f * fabsf(rx[s0] * ry[s1] - ry[s0] * rx[s1]);
        area[j] = a;
        asum += a;
    }
    if (asum == 0.f) asum = 10000.f;

    // weighted per-channel unit normals
    float gc[9];
#pragma unroll
    for (int i = 0; i < 9; ++i) gc[i] = 0.f;
#pragma unroll
    for (int j = 0; j < G; ++j) {
        const int s0 = so[j], s1 = so[(j + 1) % G];
        const float w  = area[j] / asum;
        const float x1 = rx[s0], y1 = ry[s0];
        const float x2 = rx[s1], y2 = ry[s1];
#pragma unroll
        for (int i = 0; i < 3; ++i) {
            const float c1 = rv[s0][i], c2 = rv[s1][i];
            const float nx = y1 * c2 - c1 * y2;
            const float ny = c1 * x2 - x1 * c2;
            const float nz = x1 * y2 - y1 * x2;
            const float nrm = sqrtf(nx * nx + ny * ny + nz * nz);
            const float dn  = (nrm == 0.f) ? 1.0f : nrm;
            gc[3 * i + 0] += (nx / dn) * w;
            gc[3 * i + 1] += (ny / dn) * w;
            gc[3 * i + 2] += (nz / dn) * w;
        }
    }

#pragma unroll
    for (int i = 0; i < 3; ++i) {
        const float a = gc[3 * i + 0], bb = gc[3 * i + 1], c = gc[3 * i + 2];
        const float g0 = (c == 0.f) ? 0.f : (-a / c);
        const float g1 = (c == 0.f) ? 0.f : (-bb / c);
        grad_out[(size_t)gid * 6 + 2 * i + 0] = g0 * 1e-4f;
        grad_out[(size_t)gid * 6 + 2 * i + 1] = g1 * 1e-4f;
    }

#pragma unroll
    for (int j = 0; j < G; ++j) {
        const int j1 = (j + 1) % G;
        const int s0 = so[j], s1 = so[j1];
        const float d1 = sqrtf(rx[s0] * rx[s0] + ry[s0] * ry[s0]);
        const float d2 = sqrtf(rx[s1] * rx[s1] + ry[s1] * ry[s1]);
        const float sa = fabsf(sinf((ps[j1] - ps[j] - 0.5f) * TWO_PI_F));
        float* o = umb_out + ((size_t)gid * G + j) * 28;
        o[0]  = px;        o[1]  = py;
        o[2]  = pv[0];     o[3]  = pv[1];     o[4]  = pv[2];
        o[5]  = ux[s0];    o[6]  = uy[s0];
        o[7]  = uv[s0][0]; o[8]  = uv[s0][1]; o[9]  = uv[s0][2];
        o[10] = rx[s0];    o[11] = ry[s0];
        o[12] = rv[s0][0]; o[13] = rv[s0][1]; o[14] = rv[s0][2];
        o[15] = ux[s1];    o[16] = uy[s1];
        o[17] = uv[s1][0]; o[18] = uv[s1][1]; o[19] = uv[s1][2];
        o[20] = rx[s1];    o[21] = ry[s1];
        o[22] = rv[s1][0]; o[23] = rv[s1][1]; o[24] = rv[s1][2];
        o[25] = d1;        o[26] = d2;        o[27] = sa;
        sidx_out[(size_t)gid * G + j] = (float)so[j];
    }
}

extern "C" void kernel_launch(void* const* d_in, const int* in_sizes, int n_in,
                              void* d_out, int out_size, void* d_ws, size_t ws_size,
                              hipStream_t stream) {
    const float* coord = (const float*)d_in[0];   // [B,N,2] f32
    const float* value = (const float*)d_in[1];   // [B,N,3] f32
    float* out = (float*)d_out;

    const size_t nGrad = (size_t)BATCH * NPTS * 6;
    const size_t nIdx  = (size_t)BATCH * NPTS * KNN;
    const size_t nUmb  = (size_t)BATCH * NPTS * G * 28;
    float* grad_out = out;
    float* idx_out  = out + nGrad;
    float* umb_out  = out + nGrad + nIdx;
    float* sidx_out = out + nGrad + nIdx + nUmb;

    int* idx_ws = (int*)d_ws;   // [B,N,6] int32 scratch

    knn_wmma_kernel<<<BATCH * (NPTS / 16), 32, 0, stream>>>(coord, idx_ws, idx_out);
    umbrella_kernel<<<(BATCH * NPTS + 255) / 256, 256, 0, stream>>>(
        coord, value, idx_ws, grad_out, umb_out, sidx_out);
}